// HierarchicalPolarization_75067438399945
// MI455X (gfx1250) — compile-verified
//
#include <hip/hip_runtime.h>

// HierarchicalPolarization for MI455X (gfx1250, wave32).
// Z state lives in d_out as float2 (re,im) == final (B,S,D,2) layout (134MB,
// fits in 192MB L2). Fused structure per level l:
//   [WMMA f32 MLP on level-l aggregates]  ->  [Z += At_l broadcast  AND
//   accumulate level-(l+1) weighted-reduction numerators in the same sweep]
// so Z is touched 2x per level (read+write) instead of 3x: ~3.75GB total
// traffic vs 23.3TB/s HBM (and largely L2-resident).
// MLP GEMMs (16x256 @ 256x32, 16x32 @ 32x256) use V_WMMA_F32_16X16X4_F32.

#define BB 16
#define SS 8192
#define DD 128
#define HH 32

typedef __attribute__((ext_vector_type(2))) float v2f;
typedef __attribute__((ext_vector_type(8))) float v8f;

__global__ void hp_zero_kernel(float* __restrict__ NA, float* __restrict__ denom,
                               int groups) {
    long stride = (long)gridDim.x * blockDim.x;
    long n1 = (long)groups * 256;            // numerator floats (re,im per channel)
    for (long i = (long)blockIdx.x * blockDim.x + threadIdx.x; i < n1; i += stride)
        NA[i] = 0.0f;
    for (long i = (long)blockIdx.x * blockDim.x + threadIdx.x; i < groups; i += stride)
        denom[i] = 0.0f;
}

// Interleave Z_real/Z_imag -> Z (float2) and accumulate level-0 reduction
// (groups of 2 rows: global group id = globalRow >> 1) in the same pass.
// One wave per row; lane l covers channels l, l+32, l+64, l+96.
__global__ void hp_init_reduce_kernel(const float* __restrict__ re,
                                      const float* __restrict__ im,
                                      float2* __restrict__ Z,
                                      float2* __restrict__ NA,
                                      float* __restrict__ denom) {
    int lane = threadIdx.x & 31;
    int wave = threadIdx.x >> 5;
    long row = (long)blockIdx.x * 4 + wave;          // 4 waves per block
    const float* rr = re + row * DD;
    const float* ri = im + row * DD;
    float2* zr = Z + row * DD;

    float2 z[4];
    float asum = 0.f;
#pragma unroll
    for (int k = 0; k < 4; ++k) {
        int d = lane + 32 * k;
        z[k] = make_float2(rr[d], ri[d]);
        zr[d] = z[k];
        asum += sqrtf(z[k].x * z[k].x + z[k].y * z[k].y);
    }
#pragma unroll
    for (int off = 16; off > 0; off >>= 1)
        asum += __shfl_xor(asum, off, 32);
    float mag = asum * (1.0f / 128.0f);              // mean over D

    long group = row >> 1;                           // level-0: gs = 2
    float* nbase = (float*)(NA + group * DD);
#pragma unroll
    for (int k = 0; k < 4; ++k) {
        int d = lane + 32 * k;
        atomicAdd(nbase + 2 * d,     z[k].x * mag);
        atomicAdd(nbase + 2 * d + 1, z[k].y * mag);
    }
    if (lane == 0) atomicAdd(denom + group, mag);
}

__device__ __forceinline__ float hp_gelu(float x) {
    float x3 = x * x * x;
    return 0.5f * x * (1.0f + tanhf(0.7978845608f * (x + 0.044715f * x3)));
}

// One wave (32 threads) per 16-group tile. feat(16x256) -> W1(256x32) gelu
// -> W2(32x256) -> At = (m cos p, m sin p). Both GEMMs via v_wmma_f32_16x16x4_f32.
// Fragment layouts per ISA 7.12.2:
//   A 16x4:  lane<16: M=lane,    VGPR0=K0,VGPR1=K1 ; lane>=16: M=lane-16, K2,K3
//   B 4x16:  VGPRr:   lane<16 -> row K=r+0 (half 0), lane>=16 -> row K=r+2
//   C 16x16: VGPRi:   lane<16 -> M=i, N=lane ; lane>=16 -> M=i+8, N=lane-16
__global__ void hp_block_kernel(float2* __restrict__ NA,
                                const float* __restrict__ denomBuf,
                                const float* __restrict__ W1,  // 256x32 row-major
                                const float* __restrict__ b1,  // 32
                                const float* __restrict__ W2,  // 32x256 row-major
                                const float* __restrict__ b2)  // 256
{
    __shared__ float feat[16 * 256];
    __shared__ float hbuf[16 * 32];
    __shared__ float outb[16 * 256];

    int lane = threadIdx.x;
    int row0 = blockIdx.x * 16;
    int half = lane >> 4;      // 0: lanes 0-15, 1: lanes 16-31
    int lm   = lane & 15;

    // Build feat = [|A| , angle(A)] for 16 rows, A = N / (den + eps)
    for (int idx = lane; idx < 16 * 256; idx += 32) {
        int m = idx >> 8;
        int c = idx & 255;
        int g = row0 + m;
        float den = denomBuf[g] + 1e-6f;
        float2 nv = NA[(long)g * DD + (c & 127)];
        float re = nv.x / den;
        float im = nv.y / den;
        feat[idx] = (c < 128) ? sqrtf(re * re + im * im) : atan2f(im, re);
    }
    __syncthreads();

    // GEMM1: h = gelu(feat @ W1 + b1); 2 N-tiles of 16, K=256 in 64 steps of 4
    for (int nt = 0; nt < 2; ++nt) {
        v8f c;
        float bias = b1[nt * 16 + lm];
#pragma unroll
        for (int i = 0; i < 8; ++i) c[i] = bias;
        for (int k = 0; k < 64; ++k) {
            int k0 = k * 4 + half * 2;
            v2f a, bm;
            a.x  = feat[lm * 256 + k0];
            a.y  = feat[lm * 256 + k0 + 1];
            bm.x = W1[k0 * HH + nt * 16 + lm];
            bm.y = W1[(k0 + 1) * HH + nt * 16 + lm];
            c = __builtin_amdgcn_wmma_f32_16x16x4_f32(
                    false, a, false, bm, (short)0, c, false, false);
        }
#pragma unroll
        for (int i = 0; i < 8; ++i) {
            int m = i + half * 8;
            hbuf[m * HH + nt * 16 + lm] = hp_gelu(c[i]);
        }
    }
    __syncthreads();

    // GEMM2: out = h @ W2 + b2; 16 N-tiles of 16, K=32 in 8 steps of 4
    for (int j = 0; j < 16; ++j) {
        v8f c;
        float bias = b2[j * 16 + lm];
#pragma unroll
        for (int i = 0; i < 8; ++i) c[i] = bias;
#pragma unroll
        for (int k = 0; k < 8; ++k) {
            int k0 = k * 4 + half * 2;
            v2f a, bm;
            a.x  = hbuf[lm * HH + k0];
            a.y  = hbuf[lm * HH + k0 + 1];
            bm.x = W2[k0 * 256 + j * 16 + lm];
            bm.y = W2[(k0 + 1) * 256 + j * 16 + lm];
            c = __builtin_amdgcn_wmma_f32_16x16x4_f32(
                    false, a, false, bm, (short)0, c, false, false);
        }
#pragma unroll
        for (int i = 0; i < 8; ++i) {
            int m = i + half * 8;
            outb[m * 256 + j * 16 + lm] = c[i];
        }
    }
    __syncthreads();

    // At = (m*cos p, m*sin p); overwrite numerator buffer in place (same rows)
    for (int idx = lane; idx < 16 * 128; idx += 32) {
        int m = idx >> 7;
        int d = idx & 127;
        float mg = outb[m * 256 + d];
        float ph = outb[m * 256 + 128 + d];
        NA[(long)(row0 + m) * DD + d] = make_float2(mg * cosf(ph), mg * sinf(ph));
    }
}

// Fused: Z += At_l broadcast (write back), and in the same sweep accumulate
// the level-(l+1) weighted reduction of the UPDATED Z into NA2/den2.
// Chunks partition rows of next-level groups (gs2 = 2^(shiftL+1)).
__global__ void hp_update_reduce_kernel(float2* __restrict__ Z,
                                        const float2* __restrict__ At,
                                        float2* __restrict__ NA2,
                                        float* __restrict__ den2,
                                        int shiftL,   // log2(gs_l) = level+1
                                        int ng,       // level-l groups per batch
                                        int ng2,      // next-level groups per batch
                                        int cpg2, int rpc) {
    int chunk  = blockIdx.x;
    int group2 = chunk / cpg2;               // global next-level group = b*ng2 + g2
    int cix    = chunk % cpg2;
    int b  = group2 / ng2;
    int g2 = group2 % ng2;
    int lane   = threadIdx.x & 31;
    int wave   = threadIdx.x >> 5;
    int nwaves = blockDim.x >> 5;

    int gs2 = 2 << shiftL;                   // next-level group size
    long rowBase = (long)b * SS + (long)g2 * gs2;
    int r0 = cix * rpc;
    int r1 = r0 + rpc;

    float nre[4] = {0.f, 0.f, 0.f, 0.f};
    float nim[4] = {0.f, 0.f, 0.f, 0.f};
    float dsum = 0.f;

    for (int r = r0 + wave; r < r1; r += nwaves) {
        int s = g2 * gs2 + r;                // row within batch b
        int g = s >> shiftL;                 // level-l group within batch
        const float2* arow = At + ((long)b * ng + g) * DD;
        float2* zr = Z + (rowBase + r) * DD;

        float2 z[4];
        float asum = 0.f;
#pragma unroll
        for (int k = 0; k < 4; ++k) {
            int d = lane + 32 * k;
            float2 zv = zr[d];
            float2 av = arow[d];
            zv.x += av.x;
            zv.y += av.y;
            zr[d] = zv;                      // write updated Z
            z[k] = zv;
            asum += sqrtf(zv.x * zv.x + zv.y * zv.y);
        }
#pragma unroll
        for (int off = 16; off > 0; off >>= 1)
            asum += __shfl_xor(asum, off, 32);
        float mag = asum * (1.0f / 128.0f);  // mean over D
        dsum += mag;
#pragma unroll
        for (int k = 0; k < 4; ++k) {
            nre[k] += z[k].x * mag;
            nim[k] += z[k].y * mag;
        }
    }

    float* nbase = (float*)(NA2 + (long)group2 * DD);
#pragma unroll
    for (int k = 0; k < 4; ++k) {
        int d = lane + 32 * k;
        atomicAdd(nbase + 2 * d,     nre[k]);
        atomicAdd(nbase + 2 * d + 1, nim[k]);
    }
    if (lane == 0) atomicAdd(den2 + group2, dsum);
}

// Final level: plain broadcast add (no next level to reduce for).
__global__ void hp_update_kernel(float2* __restrict__ Z,
                                 const float2* __restrict__ At,
                                 int shift, int ng, long n) {
    long stride = (long)gridDim.x * blockDim.x;
    for (long i = (long)blockIdx.x * blockDim.x + threadIdx.x; i < n; i += stride) {
        long bs = i >> 7;
        int  d  = (int)(i & 127);
        long b  = bs >> 13;                  // / SS
        int  s  = (int)(bs & (SS - 1));
        int  g  = s >> shift;
        float2 a = At[((long)b * ng + g) * DD + d];
        float2 z = Z[i];
        z.x += a.x;
        z.y += a.y;
        Z[i] = z;
    }
}

extern "C" void kernel_launch(void* const* d_in, const int* in_sizes, int n_in,
                              void* d_out, int out_size, void* d_ws, size_t ws_size,
                              hipStream_t stream) {
    const float* Z_real = (const float*)d_in[0];
    const float* Z_imag = (const float*)d_in[1];
    const float* W1 = (const float*)d_in[2];
    const float* b1 = (const float*)d_in[3];
    const float* W2 = (const float*)d_in[4];
    const float* b2 = (const float*)d_in[5];

    float2* Z = (float2*)d_out;

    // Workspace: ping-pong aggregate buffers (even levels -> A, odd -> B).
    //   bufA: 65536 groups * 128 * float2 = 64 MiB
    //   bufB: 32768 groups * 128 * float2 = 32 MiB
    //   denA: 65536 floats, denB: 32768 floats
    char* wsb = (char*)d_ws;
    float2* bufA = (float2*)wsb;
    float2* bufB = (float2*)(wsb + (size_t)65536 * DD * sizeof(float2));
    float*  denA = (float*) (wsb + (size_t)(65536 + 32768) * DD * sizeof(float2));
    float*  denB = denA + 65536;

    const long nElem = (long)BB * SS * DD;
    const long nRows = (long)BB * SS;

    // Level-0 accumulators must be zero before the fused init pass.
    hp_zero_kernel<<<4096, 256, 0, stream>>>((float*)bufA, denA, BB * (SS / 2));

    // Interleave inputs into Z and do the level-0 reduction in one sweep.
    hp_init_reduce_kernel<<<(int)(nRows / 4), 128, 0, stream>>>(
        Z_real, Z_imag, Z, bufA, denA);

    for (int level = 0; level < 13; ++level) {
        int gs = 2 << level;
        int ng = SS / gs;
        int groups = BB * ng;

        float2* cur   = (level & 1) ? bufB : bufA;
        float*  dcur  = (level & 1) ? denB : denA;

        int t = level < 9 ? level : 9;        // min(level, NT-1)
        hp_block_kernel<<<groups / 16, 32, 0, stream>>>(
            cur, dcur,
            W1 + (size_t)t * 2 * DD * HH,
            b1 + (size_t)t * HH,
            W2 + (size_t)t * HH * 2 * DD,
            b2 + (size_t)t * 2 * DD);

        if (level < 12) {
            int gs2 = gs * 2;
            int ng2 = SS / gs2;
            int groups2 = BB * ng2;
            float2* nxt  = (level & 1) ? bufA : bufB;
            float*  dnxt = (level & 1) ? denA : denB;

            int zb = groups2 > 4096 ? 4096 : groups2;
            hp_zero_kernel<<<zb, 256, 0, stream>>>((float*)nxt, dnxt, groups2);

            int rpc  = gs2 < 16 ? gs2 : 16;   // rows per chunk (within a g2 group)
            int cpg2 = gs2 / rpc;             // chunks per next-level group
            int nw   = rpc < 4 ? rpc : 4;     // waves per block
            hp_update_reduce_kernel<<<groups2 * cpg2, 32 * nw, 0, stream>>>(
                Z, cur, nxt, dnxt, level + 1, ng, ng2, cpg2, rpc);
        } else {
            hp_update_kernel<<<4096, 256, 0, stream>>>(
                Z, cur, level + 1, ng, nElem);
        }
    }
}